// Player2Vec_83760452206963
// MI455X (gfx1250) — compile-verified
//
#include <hip/hip_runtime.h>
#include <hip/hip_bf16.h>
#include <math.h>

// ---- problem constants (match reference) ----
#define NODES   50000
#define META    4
#define IN_DIM  512
#define NHID    128
#define NCLS    2
#define WEIGHT_DECAY 5e-4f
#define BN_EPS  1e-3f

#define ROW_TILES ((NODES) / 16)   // 3125
#define COL_TILES ((NHID) / 16)    // 8

typedef __attribute__((ext_vector_type(2))) float v2f;
typedef __attribute__((ext_vector_type(8))) float v8f;

// ---------------------------------------------------------------------------
// zero fill
// ---------------------------------------------------------------------------
__global__ void zero_kernel(float* __restrict__ p, size_t count) {
    size_t stride = (size_t)gridDim.x * blockDim.x;
    for (size_t i = (size_t)blockIdx.x * blockDim.x + threadIdx.x; i < count; i += stride)
        p[i] = 0.0f;
}

// ---------------------------------------------------------------------------
// GEMM1: pre1[50000,128] = x[50000,512] @ W1[512,128]
// fp32 WMMA 16x16x4. One wave owns a full 16-row x 128-col tile using 8
// independent accumulators: each A fragment is loaded from HBM exactly once
// (x read once total = 102.4 MB), and the 8 WMMAs per k-step are mutually
// independent so they pipeline back-to-back on the matrix unit. W1 (256 KB)
// is L2-resident, so its repeated fragment loads are cache hits.
//
// A 16x4 layout: lanes 0-15 -> {K=0,K=1}, lanes 16-31 -> {K=2,K=3}
// B 4x16 layout: lanes 0-15 -> {K=0,K=1} at N=lane, lanes 16-31 -> {K=2,K=3}
// C/D 16x16:     VGPR r: lanes 0-15 -> M=r, lanes 16-31 -> M=r+8
// ---------------------------------------------------------------------------
__global__ void gemm1_wmma(const float* __restrict__ x,
                           const float* __restrict__ W1,
                           float* __restrict__ pre1) {
    const int lane = threadIdx.x & 31;
    const int wave = threadIdx.x >> 5;
    const int t    = blockIdx.x * 8 + wave;    // row tile index
    if (t >= ROW_TILES) return;
    const int m0 = t * 16;
    const int hi = lane >> 4;                  // 0: lanes 0-15, 1: lanes 16-31
    const int lm = lane & 15;

    const float* arow = x + (size_t)(m0 + lm) * IN_DIM;

    v8f acc[COL_TILES];
#pragma unroll
    for (int ct = 0; ct < COL_TILES; ++ct) acc[ct] = (v8f){};

    for (int k = 0; k < IN_DIM; k += 4) {
        const int ka = k + 2 * hi;
        v2f a;
        a.x = arow[ka];
        a.y = arow[ka + 1];
        const float* w0 = W1 + (size_t)ka * NHID + lm;
#pragma unroll
        for (int ct = 0; ct < COL_TILES; ++ct) {
            v2f b;
            b.x = w0[ct * 16];
            b.y = w0[NHID + ct * 16];
            acc[ct] = __builtin_amdgcn_wmma_f32_16x16x4_f32(
                /*neg_a=*/false, a, /*neg_b=*/false, b,
                /*c_mod=*/(short)0, acc[ct], /*reuse_a=*/false, /*reuse_b=*/false);
        }
    }

#pragma unroll
    for (int ct = 0; ct < COL_TILES; ++ct) {
#pragma unroll
        for (int r = 0; r < 8; ++r) {
            const int row = m0 + r + 8 * hi;
            pre1[(size_t)row * NHID + ct * 16 + lm] = acc[ct][r];
        }
    }
}

// ---------------------------------------------------------------------------
// SpMM layer 1: h1[m][r][:] += v * pre1[c][:]  (one wave per edge, 128 feats)
// ---------------------------------------------------------------------------
__global__ void spmm1_atomic(const int* __restrict__ rows,
                             const int* __restrict__ cols,
                             const float* __restrict__ vals,
                             const float* __restrict__ pre1,
                             float* __restrict__ h1, int E) {
    const int widx = (int)(((size_t)blockIdx.x * blockDim.x + threadIdx.x) >> 5);
    const int lane = threadIdx.x & 31;
    if (widx >= META * E) return;
    const int meta = widx / E;
    const int e    = widx - meta * E;
    const size_t eg = (size_t)meta * E + e;
    const int r = rows[eg];
    const int c = cols[eg];
    const float v = vals[eg];
    const float* src = pre1 + (size_t)c * NHID;
    float* dst = h1 + ((size_t)meta * NODES + r) * NHID;
#pragma unroll
    for (int j = 0; j < 4; ++j) {
        const int f = lane + 32 * j;
        atomicAdd(&dst[f], v * src[f]);
    }
}

// ---------------------------------------------------------------------------
// BN statistics over node axis: one block per (meta, feature) column
// stats[0..511] = mean, stats[512..1023] = rsqrt(var + eps)
// ---------------------------------------------------------------------------
__global__ void bn_stats(const float* __restrict__ h1, float* __restrict__ stats) {
    const int col  = blockIdx.x;          // 0..META*NHID-1
    const int meta = col / NHID;
    const int f    = col - meta * NHID;
    const float* base = h1 + (size_t)meta * NODES * NHID + f;
    float s = 0.0f, s2 = 0.0f;
    for (int i = threadIdx.x; i < NODES; i += blockDim.x) {
        const float v = base[(size_t)i * NHID];
        s += v; s2 += v * v;
    }
    __shared__ float sh[256], sh2[256];
    sh[threadIdx.x] = s; sh2[threadIdx.x] = s2;
    __syncthreads();
    for (int o = blockDim.x >> 1; o > 0; o >>= 1) {
        if ((int)threadIdx.x < o) {
            sh[threadIdx.x]  += sh[threadIdx.x + o];
            sh2[threadIdx.x] += sh2[threadIdx.x + o];
        }
        __syncthreads();
    }
    if (threadIdx.x == 0) {
        const float mean = sh[0] / (float)NODES;
        const float var  = sh2[0] / (float)NODES - mean * mean;
        stats[col] = mean;
        stats[META * NHID + col] = rsqrtf(var + BN_EPS);
    }
}

// ---------------------------------------------------------------------------
// Fused BN + ReLU + GEMM2 (128 -> 2): one wave per (meta,node)
// ---------------------------------------------------------------------------
__global__ void bn_relu_gemm2(const float* __restrict__ h1,
                              const float* __restrict__ stats,
                              const float* __restrict__ W2,
                              float* __restrict__ pre2) {
    const int widx = (int)(((size_t)blockIdx.x * blockDim.x + threadIdx.x) >> 5);
    const int lane = threadIdx.x & 31;
    if (widx >= META * NODES) return;
    const int meta = widx / NODES;
    const int node = widx - meta * NODES;
    const float* hrow = h1 + ((size_t)meta * NODES + node) * NHID;
    float a0 = 0.0f, a1 = 0.0f;
#pragma unroll
    for (int j = 0; j < 4; ++j) {
        const int f = lane * 4 + j;
        const float mean = stats[meta * NHID + f];
        const float rstd = stats[META * NHID + meta * NHID + f];
        float hn = (hrow[f] - mean) * rstd;
        hn = fmaxf(hn, 0.0f);
        a0 += hn * W2[f * NCLS + 0];
        a1 += hn * W2[f * NCLS + 1];
    }
    for (int o = 16; o > 0; o >>= 1) {
        a0 += __shfl_xor(a0, o, 32);
        a1 += __shfl_xor(a1, o, 32);
    }
    if (lane == 0) {
        float* d = pre2 + ((size_t)meta * NODES + node) * NCLS;
        d[0] = a0; d[1] = a1;
    }
}

// ---------------------------------------------------------------------------
// SpMM layer 2 (C=2): one thread per edge
// ---------------------------------------------------------------------------
__global__ void spmm2_atomic(const int* __restrict__ rows,
                             const int* __restrict__ cols,
                             const float* __restrict__ vals,
                             const float* __restrict__ pre2,
                             float* __restrict__ h2, int E) {
    const int idx = (int)((size_t)blockIdx.x * blockDim.x + threadIdx.x);
    if (idx >= META * E) return;
    const int meta = idx / E;
    const int e    = idx - meta * E;
    const size_t eg = (size_t)meta * E + e;
    const int r = rows[eg];
    const int c = cols[eg];
    const float v = vals[eg];
    const float* s = pre2 + ((size_t)meta * NODES + c) * NCLS;
    float* d = h2 + ((size_t)meta * NODES + r) * NCLS;
    atomicAdd(&d[0], v * s[0]);
    atomicAdd(&d[1], v * s[1]);
}

// ---------------------------------------------------------------------------
// Attention partial: vacc[m][j] += sum_i h2flat[m][i] * w_omega[i][j]
// ---------------------------------------------------------------------------
__global__ void attn_partial(const float* __restrict__ h2,
                             const float* __restrict__ w_omega,
                             float* __restrict__ vacc) {
    const int FLAT = NODES * NCLS;
    float acc[META][META] = {};
    const int stride = gridDim.x * blockDim.x;
    for (int i = blockIdx.x * blockDim.x + threadIdx.x; i < FLAT; i += stride) {
        float w[META];
#pragma unroll
        for (int j = 0; j < META; ++j) w[j] = w_omega[(size_t)i * META + j];
#pragma unroll
        for (int m = 0; m < META; ++m) {
            const float fm = h2[(size_t)m * FLAT + i];
#pragma unroll
            for (int j = 0; j < META; ++j) acc[m][j] += fm * w[j];
        }
    }
    __shared__ float sh[META * META];
    if (threadIdx.x < META * META) sh[threadIdx.x] = 0.0f;
    __syncthreads();
#pragma unroll
    for (int m = 0; m < META; ++m)
#pragma unroll
        for (int j = 0; j < META; ++j)
            atomicAdd(&sh[m * META + j], acc[m][j]);
    __syncthreads();
    if (threadIdx.x < META * META)
        atomicAdd(&vacc[threadIdx.x], sh[threadIdx.x]);
}

// ---------------------------------------------------------------------------
// Attention finalize: v = tanh(vacc + b); s = v @ u; alphas = softmax(s)
// ---------------------------------------------------------------------------
__global__ void attn_finalize(const float* __restrict__ vacc,
                              const float* __restrict__ b_omega,
                              const float* __restrict__ u_omega,
                              float* __restrict__ alphas) {
    float s[META];
    float mx = -1e30f;
    for (int m = 0; m < META; ++m) {
        float sm = 0.0f;
        for (int j = 0; j < META; ++j)
            sm += tanhf(vacc[m * META + j] + b_omega[j]) * u_omega[j];
        s[m] = sm;
        mx = fmaxf(mx, sm);
    }
    float den = 0.0f, ex[META];
    for (int m = 0; m < META; ++m) { ex[m] = expf(s[m] - mx); den += ex[m]; }
    for (int m = 0; m < META; ++m) alphas[m] = ex[m] / den;
}

// ---------------------------------------------------------------------------
// generic reductions -> atomicAdd into a scalar
// ---------------------------------------------------------------------------
__global__ void reduce_sq(const float* __restrict__ p, int count, float* __restrict__ out) {
    float s = 0.0f;
    const int stride = gridDim.x * blockDim.x;
    for (int i = blockIdx.x * blockDim.x + threadIdx.x; i < count; i += stride)
        s += p[i] * p[i];
    __shared__ float sh[256];
    sh[threadIdx.x] = s;
    __syncthreads();
    for (int o = blockDim.x >> 1; o > 0; o >>= 1) {
        if ((int)threadIdx.x < o) sh[threadIdx.x] += sh[threadIdx.x + o];
        __syncthreads();
    }
    if (threadIdx.x == 0) atomicAdd(out, sh[0]);
}

__global__ void reduce_sum(const float* __restrict__ p, int count, float* __restrict__ out) {
    float s = 0.0f;
    const int stride = gridDim.x * blockDim.x;
    for (int i = blockIdx.x * blockDim.x + threadIdx.x; i < count; i += stride)
        s += p[i];
    __shared__ float sh[256];
    sh[threadIdx.x] = s;
    __syncthreads();
    for (int o = blockDim.x >> 1; o > 0; o >>= 1) {
        if ((int)threadIdx.x < o) sh[threadIdx.x] += sh[threadIdx.x + o];
        __syncthreads();
    }
    if (threadIdx.x == 0) atomicAdd(out, sh[0]);
}

// ---------------------------------------------------------------------------
// per-node CE + accuracy: scal[2]+=sum ce_i*mask_i, scal[3]+=sum hit_i*mask_i
// ---------------------------------------------------------------------------
__global__ void loss_kernel(const float* __restrict__ h2,
                            const float* __restrict__ alphas,
                            const float* __restrict__ label,
                            const float* __restrict__ mask,
                            float* __restrict__ scal) {
    const int i = blockIdx.x * blockDim.x + threadIdx.x;
    float ce = 0.0f, ac = 0.0f;
    if (i < NODES) {
        float l0 = 0.0f, l1 = 0.0f;
#pragma unroll
        for (int m = 0; m < META; ++m) {
            const float a = alphas[m];
            const float* hm = h2 + (size_t)m * NODES * NCLS + (size_t)i * NCLS;
            l0 += a * hm[0];
            l1 += a * hm[1];
        }
        const float mx = fmaxf(l0, l1);
        const float z  = mx + logf(expf(l0 - mx) + expf(l1 - mx));
        const float ls0 = l0 - z, ls1 = l1 - z;
        const float lb0 = label[(size_t)i * NCLS + 0];
        const float lb1 = label[(size_t)i * NCLS + 1];
        const float mi  = mask[i];
        ce = -(lb0 * ls0 + lb1 * ls1) * mi;
        const int pred = (l1 > l0) ? 1 : 0;
        const int lab  = (lb1 > lb0) ? 1 : 0;
        ac = (pred == lab ? 1.0f : 0.0f) * mi;
    }
    __shared__ float shc[256], sha[256];
    shc[threadIdx.x] = ce; sha[threadIdx.x] = ac;
    __syncthreads();
    for (int o = blockDim.x >> 1; o > 0; o >>= 1) {
        if ((int)threadIdx.x < o) {
            shc[threadIdx.x] += shc[threadIdx.x + o];
            sha[threadIdx.x] += sha[threadIdx.x + o];
        }
        __syncthreads();
    }
    if (threadIdx.x == 0) {
        atomicAdd(&scal[2], shc[0]);
        atomicAdd(&scal[3], sha[0]);
    }
}

// scal: [0]=sum(var^2) raw, [1]=sum(mask), [2]=sum ce*mask, [3]=sum hit*mask
__global__ void final_kernel(const float* __restrict__ scal, float* __restrict__ out) {
    const float l2   = 0.5f * scal[0];
    const float msum = scal[1];
    out[0] = WEIGHT_DECAY * l2 + scal[2] / msum;   // loss
    out[1] = scal[3] / msum;                       // acc
}

// ---------------------------------------------------------------------------
extern "C" void kernel_launch(void* const* d_in, const int* in_sizes, int n_in,
                              void* d_out, int out_size, void* d_ws, size_t ws_size,
                              hipStream_t stream) {
    const float* x       = (const float*)d_in[0];
    const float* W1      = (const float*)d_in[1];
    const float* W2      = (const float*)d_in[2];
    const float* w_omega = (const float*)d_in[3];
    const float* b_omega = (const float*)d_in[4];
    const float* u_omega = (const float*)d_in[5];
    const int*   sup_rows = (const int*)d_in[6];
    const int*   sup_cols = (const int*)d_in[7];
    const float* sup_vals = (const float*)d_in[8];
    const float* label   = (const float*)d_in[9];
    const float* mask    = (const float*)d_in[10];

    const int E = in_sizes[6] / META;   // 800000

    float* ws = (float*)d_ws;
    float* pre1   = ws;                                           // NODES*NHID
    float* h1     = pre1  + (size_t)NODES * NHID;                 // META*NODES*NHID
    float* stats  = h1    + (size_t)META * NODES * NHID;          // 2*META*NHID
    float* pre2   = stats + 2 * META * NHID;                      // META*NODES*NCLS
    float* h2     = pre2  + (size_t)META * NODES * NCLS;          // META*NODES*NCLS
    float* vacc   = h2    + (size_t)META * NODES * NCLS;          // 16
    float* alphas = vacc  + META * META;                          // 4
    float* scal   = alphas + META;                                // 4

    // zero accumulation buffers (h1; then h2+vacc+alphas+scal contiguous)
    zero_kernel<<<2048, 256, 0, stream>>>(h1, (size_t)META * NODES * NHID);
    zero_kernel<<<1024, 256, 0, stream>>>(h2, (size_t)META * NODES * NCLS
                                              + META * META + META + 4);

    // layer 1 dense GEMM (WMMA fp32): 3125 row tiles, 8 waves/block
    gemm1_wmma<<<(ROW_TILES + 7) / 8, 256, 0, stream>>>(x, W1, pre1);

    // layer 1 sparse propagation
    {
        const size_t threads = (size_t)META * E * 32;
        spmm1_atomic<<<(unsigned)((threads + 255) / 256), 256, 0, stream>>>(
            sup_rows, sup_cols, sup_vals, pre1, h1, E);
    }

    // batch-norm stats + fused BN/ReLU/GEMM2
    bn_stats<<<META * NHID, 256, 0, stream>>>(h1, stats);
    bn_relu_gemm2<<<(META * NODES + 7) / 8, 256, 0, stream>>>(h1, stats, W2, pre2);

    // layer 2 sparse propagation
    spmm2_atomic<<<(META * E + 255) / 256, 256, 0, stream>>>(
        sup_rows, sup_cols, sup_vals, pre2, h2, E);

    // attention fusion
    attn_partial<<<512, 256, 0, stream>>>(h2, w_omega, vacc);
    attn_finalize<<<1, 1, 0, stream>>>(vacc, b_omega, u_omega, alphas);

    // L2 regularization sums + mask sum
    reduce_sq<<<128, 256, 0, stream>>>(W1, IN_DIM * NHID, &scal[0]);
    reduce_sq<<<1, 256, 0, stream>>>(W2, NHID * NCLS, &scal[0]);
    reduce_sq<<<256, 256, 0, stream>>>(w_omega, in_sizes[3], &scal[0]);
    reduce_sq<<<1, 256, 0, stream>>>(b_omega, META, &scal[0]);
    reduce_sq<<<1, 256, 0, stream>>>(u_omega, META, &scal[0]);
    reduce_sum<<<128, 256, 0, stream>>>(mask, NODES, &scal[1]);

    // masked CE + accuracy, then finalize two scalars
    loss_kernel<<<(NODES + 255) / 256, 256, 0, stream>>>(h2, alphas, label, mask, scal);
    final_kernel<<<1, 1, 0, stream>>>(scal, (float*)d_out);
}